// RandomizedCollider_77876347011412
// MI455X (gfx1250) — compile-verified
//
#include <hip/hip_runtime.h>
#include <stdint.h>

typedef float v2f __attribute__((ext_vector_type(2)));
typedef float v8f __attribute__((ext_vector_type(8)));

#define N_PART 2048
#define TILE 16
#define NTILES (N_PART / TILE)   // 128 row tiles
#define RESTITUTION 0.5f
#define EPS 1e-12f

// splitmix-style hash over (seed, i, j) for the bernoulli gate + reservoir key
__device__ __forceinline__ uint32_t hash3(uint32_t a, uint32_t b, uint32_t c) {
  uint32_t h = a * 0x9E3779B1u;
  h ^= b + 0x85EBCA77u + (h << 6) + (h >> 2);
  h *= 0xC2B2AE3Du;
  h ^= c + 0x27D4EB2Fu + (h << 6) + (h >> 2);
  h ^= h >> 16; h *= 0x7FEB352Du;
  h ^= h >> 15; h *= 0x846CA68Bu;
  h ^= h >> 16;
  return h;
}

// native v_sqrt_f32 (~1 ULP): used only where the result feeds a threshold
// test, not the final output values.
__device__ __forceinline__ float fast_sqrtf(float x) {
  return __builtin_amdgcn_sqrtf(x);
}

// Kernel 1: one wave32 per 16-row tile. WMMA f32 16x16x4 computes the Gram
// matrix pos_i . pos_j for a 16x16 tile; dist^2 = |pi|^2 + |pj|^2 - 2*G.
// Per-row uniform random selection among kept contacts via min-key reservoir.
__global__ __launch_bounds__(32)
void collider_pairs_kernel(const float* __restrict__ pos,
                           const float* __restrict__ rad,
                           const int*  __restrict__ seedp,
                           int*   __restrict__ chosen_j,   // [N], -1 = none
                           float* __restrict__ pen_sel)    // [N*2]
{
  const int lane = threadIdx.x;            // 0..31
  const int ti   = blockIdx.x;             // row tile index
  const int i0   = ti * TILE;
  const int g    = lane >> 4;              // half-wave: 0 -> rows 0..7, 1 -> rows 8..15
  const int l    = lane & 15;
  const uint32_t seed = (uint32_t)seedp[0];

  // A operand (16x4, K = x,y,0,0): lanes 0-15 hold (x,y) for row i0+l, lanes 16-31 pad
  float arx = pos[2 * (i0 + l) + 0];
  float ary = pos[2 * (i0 + l) + 1];
  v2f a;
  a.x = (g == 0) ? arx : 0.0f;
  a.y = (g == 0) ? ary : 0.0f;

  // per-row-slot constants: rows i = i0 + 8*g + r
  float ri2[8], radi[8];
#pragma unroll
  for (int r = 0; r < 8; ++r) {
    int i = i0 + 8 * g + r;
    float px = pos[2 * i], py = pos[2 * i + 1];
    ri2[r]  = px * px + py * py;
    radi[r] = rad[i];
  }

  float minkey[8];
  int   minj[8];
  int   cnt[8];
#pragma unroll
  for (int r = 0; r < 8; ++r) { minkey[r] = 3.0e38f; minj[r] = -1; cnt[r] = 0; }

  // lower-triangle column tiles only (valid requires i > j)
  for (int tj = 0; tj <= ti; ++tj) {
    const int j0 = tj * TILE;
    const int j  = j0 + l;                  // column this lane handles in C tile

    // prefetch next column tile (global_prefetch_b8) behind this tile's work
    if (tj < ti) {
      __builtin_prefetch(&pos[2 * (j + TILE)], 0, 1);
    }

    float bx = pos[2 * j], by = pos[2 * j + 1];
    float rj  = rad[j];
    float rj2 = bx * bx + by * by;

    // B operand (4x16, K = x,y,0,0): lanes 0-15 hold (x,y) of col j0+l, lanes 16-31 pad
    v2f b;
    b.x = (g == 0) ? bx : 0.0f;
    b.y = (g == 0) ? by : 0.0f;

    v8f c = {};
    // D = A x B: c[r] = pos_i . pos_j for row m = 8*g + r, col n = l  (EXEC all 1s)
    c = __builtin_amdgcn_wmma_f32_16x16x4_f32(
        /*neg_a=*/false, a, /*neg_b=*/false, b,
        /*c_mod=*/(short)0, c, /*reuse_a=*/false, /*reuse_b=*/false);

#pragma unroll
    for (int r = 0; r < 8; ++r) {
      int i = i0 + 8 * g + r;
      float d2   = ri2[r] + rj2 - 2.0f * c[r];
      float dist = fast_sqrtf(fmaxf(d2, 0.0f) + EPS);   // threshold only -> native sqrt
      float pen  = radi[r] + rj - dist;
      bool valid = (pen > 0.0f) && (i > j);
      uint32_t h = hash3(seed, (uint32_t)i, (uint32_t)j);
      bool keep  = valid && (h & 1u);                     // bernoulli(0.5) gate
      float key  = (float)(h >> 8) * (1.0f / 16777216.0f);
      bool better = keep && (key < minkey[r]);
      minkey[r] = better ? key : minkey[r];
      minj[r]   = better ? j   : minj[r];
      cnt[r]   += keep ? 1 : 0;
    }
  }

  // reduce reservoirs across the 16 lanes of each half-wave (rows live per-half)
#pragma unroll
  for (int r = 0; r < 8; ++r) {
    for (int m = 1; m < 16; m <<= 1) {
      float ok = __shfl_xor(minkey[r], m, 16);
      int   oj = __shfl_xor(minj[r],   m, 16);
      int   oc = __shfl_xor(cnt[r],    m, 16);
      cnt[r] += oc;
      bool take = (ok < minkey[r]) || ((ok == minkey[r]) && (oj >= 0) && (oj < minj[r]));
      minkey[r] = take ? ok : minkey[r];
      minj[r]   = take ? oj : minj[r];
    }
  }

  if (l == 0) {
#pragma unroll
    for (int r = 0; r < 8; ++r) {
      int i = i0 + 8 * g + r;
      int jsel = (cnt[r] > 0) ? minj[r] : -1;
      chosen_j[i] = jsel;
      float pvx = 0.0f, pvy = 0.0f;
      if (jsel >= 0) {
        // recompute pen_vec = (d / dist) * pen with IEEE sqrt (feeds the output)
        float dx = pos[2 * i]     - pos[2 * jsel];
        float dy = pos[2 * i + 1] - pos[2 * jsel + 1];
        float dist = __builtin_sqrtf(dx * dx + dy * dy + EPS);
        float pen  = rad[i] + rad[jsel] - dist;
        pvx = dx / dist * pen;
        pvy = dy / dist * pen;
      }
      pen_sel[2 * i]     = pvx;
      pen_sel[2 * i + 1] = pvy;
    }
  }
}

// Kernel 2: single workgroup; sequential 2048-step resolution scan in LDS.
__global__ __launch_bounds__(256)
void collider_resolve_kernel(const float* __restrict__ pos,
                             const float* __restrict__ vel,
                             const float* __restrict__ mass,
                             const int*   __restrict__ chosen_j,
                             const float* __restrict__ pen_sel,
                             float* __restrict__ out)
{
  __shared__ float sp[N_PART * 2];
  __shared__ float sv[N_PART * 2];
  __shared__ float sim[N_PART];
  const int t = threadIdx.x;

  for (int i = t; i < N_PART; i += 256) {
    sp[2 * i]     = pos[2 * i];
    sp[2 * i + 1] = pos[2 * i + 1];
    sv[2 * i]     = vel[2 * i];
    sv[2 * i + 1] = vel[2 * i + 1];
    sim[i]        = 1.0f / mass[i];
  }
  __syncthreads();

  if (t == 0) {
    for (int i = 0; i < N_PART; ++i) {
      int j = chosen_j[i];
      if (j < 0 || j == i) continue;
      float pvx = pen_sel[2 * i], pvy = pen_sel[2 * i + 1];
      float imi = sim[i], imj = sim[j];
      float inv_sum = 1.0f / (imi + imj);
      float nlen = __builtin_sqrtf(pvx * pvx + pvy * pvy + EPS);
      float nx = pvx / nlen, ny = pvy / nlen;
      float corrx = pvx * inv_sum, corry = pvy * inv_sum;
      sp[2 * i]     += corrx * imi;  sp[2 * i + 1] += corry * imi;
      sp[2 * j]     -= corrx * imj;  sp[2 * j + 1] -= corry * imj;
      float vrelx = sv[2 * i] - sv[2 * j];
      float vrely = sv[2 * i + 1] - sv[2 * j + 1];
      float vrel = vrelx * nx + vrely * ny;
      float jimp = (vrel < 0.0f) ? (-(1.0f + RESTITUTION) * vrel * inv_sum) : 0.0f;
      sv[2 * i]     += jimp * imi * nx;  sv[2 * i + 1] += jimp * imi * ny;
      sv[2 * j]     -= jimp * imj * nx;  sv[2 * j + 1] -= jimp * imj * ny;
    }
  }
  __syncthreads();

  for (int i = t; i < N_PART; i += 256) {
    out[4 * i + 0] = sp[2 * i];
    out[4 * i + 1] = sp[2 * i + 1];
    out[4 * i + 2] = sv[2 * i];
    out[4 * i + 3] = sv[2 * i + 1];
  }
}

extern "C" void kernel_launch(void* const* d_in, const int* in_sizes, int n_in,
                              void* d_out, int out_size, void* d_ws, size_t ws_size,
                              hipStream_t stream) {
  const float* pos  = (const float*)d_in[0];
  const float* rad  = (const float*)d_in[1];
  const float* vel  = (const float*)d_in[2];
  const float* mass = (const float*)d_in[3];
  const int*   seed = (const int*)d_in[4];

  int*   chosen = (int*)d_ws;                                   // N ints
  float* pensel = (float*)((char*)d_ws + N_PART * sizeof(int)); // N*2 floats

  collider_pairs_kernel<<<NTILES, 32, 0, stream>>>(pos, rad, seed, chosen, pensel);
  collider_resolve_kernel<<<1, 256, 0, stream>>>(pos, vel, mass, chosen, pensel,
                                                 (float*)d_out);
}